// ConvCaps_35656818491744
// MI455X (gfx1250) — compile-verified
//
#include <hip/hip_runtime.h>

#define EPSF     1e-7f
#define VAR_EPSF 0.01f
#define NB   8
#define HH   14
#define WW   14
#define NC   32
#define PD   16
#define OHW  6
#define INC  288
#define ONC  32

typedef float v2f  __attribute__((ext_vector_type(2)));
typedef float v8f  __attribute__((ext_vector_type(8)));
typedef unsigned int u32x4 __attribute__((ext_vector_type(4)));
typedef int   i32x8 __attribute__((ext_vector_type(8)));
typedef int   i32x4 __attribute__((ext_vector_type(4)));

__device__ __forceinline__ unsigned lds_off(const void* p) {
  return (unsigned)(unsigned long long)(p);
}

// Build a 3D-tile Tensor-DMA descriptor (D#) per CDNA5 ISA ch.8 and issue
// TENSOR_LOAD_TO_LDS. dim0 = contiguous elements (f32), dim1/dim2 strided.
__device__ __forceinline__ void tdm_load_tile(
    unsigned ldsOff, unsigned long long gaddr,
    unsigned tdim0, unsigned tdim1, unsigned tdim2,
    unsigned tile0, unsigned tile1, unsigned tile2,
    unsigned long long stride0, unsigned long long stride1) {
  u32x4 g0;
  g0[0] = 1u;                                            // count=1, user mode
  g0[1] = ldsOff;                                        // lds_addr
  g0[2] = (unsigned)(gaddr & 0xFFFFFFFFull);             // global_addr[31:0]
  g0[3] = (unsigned)((gaddr >> 32) & 0x1FFFFFFull) | (2u << 30); // [56:32] | type=2
  i32x8 g1;
  g1[0] = (int)(2u << 16);                               // data_size = 4B
  g1[1] = (int)((tdim0 & 0xFFFFu) << 16);                // tensor_dim0[15:0]
  g1[2] = (int)(((tdim0 >> 16) & 0xFFFFu) | ((tdim1 & 0xFFFFu) << 16));
  g1[3] = (int)(((tdim1 >> 16) & 0xFFFFu) | ((tile0 & 0xFFFFu) << 16));
  g1[4] = (int)((tile1 & 0xFFFFu) | ((tile2 & 0xFFFFu) << 16));
  g1[5] = (int)(stride0 & 0xFFFFFFFFull);                // tensor_dim0_stride
  g1[6] = (int)(((stride0 >> 32) & 0xFFFFu) | ((stride1 & 0xFFFFu) << 16));
  g1[7] = (int)((stride1 >> 16) & 0xFFFFFFFFull);        // dim1_stride[47:16]
  i32x4 g2; g2[0] = (int)tdim2; g2[1] = 0; g2[2] = 0; g2[3] = 0;
  i32x4 g3; g3[0] = 0; g3[1] = 0; g3[2] = 0; g3[3] = 0;
  i32x8 g4; g4[0] = 0; g4[1] = 0; g4[2] = 0; g4[3] = 0;
  g4[4] = 0; g4[5] = 0; g4[6] = 0; g4[7] = 0;            // unused extra group
  __builtin_amdgcn_tensor_load_to_lds(g0, g1, g2, g3, g4, 0);
}

__global__ __launch_bounds__(256)
void convcaps_em_kernel(const float* __restrict__ pose,
                        const float* __restrict__ active,
                        const float* __restrict__ w,
                        const float* __restrict__ bv,
                        const float* __restrict__ ba,
                        float* __restrict__ out) {
  // LDS layout. x_s[ij][c]: pose tile per TDM order (c fastest, then j, i).
  // TF scramble: flat g = i*16+d maps to channel c = g/9, patch ij = g%9.
  __shared__ float x_s[9 * 512];
  __shared__ float a_s[9 * NC];
  __shared__ float rp_s[INC * ONC];   // r (post-softmax) * active[i]
  __shared__ float mu_s[ONC * PD];
  __shared__ float sig_s[ONC * PD];
  __shared__ float red_s[ONC * PD];
  __shared__ float w_s[8 * 544];      // per-wave staged w row, x17 pad
  __shared__ float rsum_s[ONC];
  __shared__ float actp_s[ONC];
  __shared__ float p0_s[ONC];

  const int t    = threadIdx.x;
  const int wave = t >> 5;
  const int lane = t & 31;
  const int pos  = blockIdx.x;
  const int b    = pos / (OHW * OHW);
  const int rem  = pos % (OHW * OHW);
  const int oy   = rem / OHW;
  const int ox   = rem % OHW;

  if ((int)blockIdx.x < 0) { x_s[0] = 0.f; a_s[0] = 0.f; }  // keep LDS stores alive

  if (wave == 0) {
    // pose tile: 512 contiguous ch, 3 cols (stride 512), 3 rows (stride 14*512)
    unsigned long long pbase = (unsigned long long)(uintptr_t)pose +
        4ull * (unsigned long long)(((b * HH + oy * 2) * WW + ox * 2) * 512);
    tdm_load_tile(lds_off(x_s), pbase, 512u, (unsigned)WW, (unsigned)HH,
                  512u, 3u, 3u, 512ull, (unsigned long long)(WW * 512));
    // activation tile: 32 contiguous ch, 3 cols (stride 32), 3 rows (stride 448)
    unsigned long long abase = (unsigned long long)(uintptr_t)active +
        4ull * (unsigned long long)(((b * HH + oy * 2) * WW + ox * 2) * NC);
    tdm_load_tile(lds_off(a_s), abase, (unsigned)NC, (unsigned)WW, (unsigned)HH,
                  (unsigned)NC, 3u, 3u, (unsigned long long)NC,
                  (unsigned long long)(WW * NC));
    __builtin_amdgcn_s_wait_tensorcnt(0);
  }
  __syncthreads();

  // r init: uniform 1/ONC, folded with active[i] (m_step does r*active)
  for (int m = t; m < INC * ONC; m += 256) {
    int i = m >> 5;
    rp_s[m] = a_s[(i % 9) * NC + (i / 9)] * (1.0f / ONC);
  }
  __syncthreads();

  const float lam_arr[3] = {0.0005f, 0.000975f, 0.00142625f};
  const int d  = t & 15;   // pose dim
  const int og = t >> 4;   // output capsule group (0..15); handles og and og+16

  for (int it = 0; it < 3; ++it) {
    // ---- r_sum[o] = sum_i rp[i,o] via WMMA ones-trick (16x16x4 f32) ----
    if (t < ONC) rsum_s[t] = 0.0f;
    __syncthreads();
    {
      const int o16 = lane & 15;
      v2f ones; ones.x = 1.0f; ones.y = 1.0f;
      for (int half = 0; half < 2; ++half) {
        const int o = o16 + 16 * half;
        v8f acc = {0.f, 0.f, 0.f, 0.f, 0.f, 0.f, 0.f, 0.f};
#pragma unroll
        for (int k = 0; k < 9; ++k) {
          int ia = wave * 36 + k * 4 + ((lane < 16) ? 0 : 2);
          v2f bb;
          bb.x = rp_s[ia * ONC + o];
          bb.y = rp_s[(ia + 1) * ONC + o];
          // D[m,o] += sum_k B[k,o]; A==ones makes result K-layout invariant
          acc = __builtin_amdgcn_wmma_f32_16x16x4_f32(
              false, ones, false, bb, (short)0, acc, false, false);
        }
        if (lane < 16) atomicAdd(&rsum_s[o], acc[0]);
      }
    }
    __syncthreads();

    // ---- m_step moments: S1 = sum r*v, S2 = sum r*v^2 (v = w*x) ----
    float s1a = 0.f, s2a = 0.f, s1b = 0.f, s2b = 0.f;
#pragma unroll 4
    for (int i = 0; i < INC; ++i) {
      int g  = i * PD + d;
      int c  = (int)((unsigned)g / 9u);
      int ij = g - c * 9;
      float x  = x_s[ij * 512 + c];
      float wa = w[(i * ONC + og) * PD + d];
      float wb = w[(i * ONC + og + 16) * PD + d];
      float ra = rp_s[i * ONC + og];
      float rb = rp_s[i * ONC + og + 16];
      float va = wa * x, vb = wb * x;
      s1a += ra * va; s2a += ra * va * va;
      s1b += rb * vb; s2b += rb * vb * vb;
    }
    {
      float S0a = rsum_s[og], S0b = rsum_s[og + 16];
      float mua = s1a / (S0a + EPSF);
      float mub = s1b / (S0b + EPSF);
      float vara = (s2a - 2.f * mua * s1a + mua * mua * S0a) / (S0a + EPSF) + VAR_EPSF;
      float varb = (s2b - 2.f * mub * s1b + mub * mub * S0b) / (S0b + EPSF) + VAR_EPSF;
      float sga = sqrtf(vara), sgb = sqrtf(varb);
      mu_s[og * PD + d]         = mua;  mu_s[(og + 16) * PD + d]  = mub;
      sig_s[og * PD + d]        = sga;  sig_s[(og + 16) * PD + d] = sgb;
      red_s[og * PD + d]        = __logf(sga + EPSF);
      red_s[(og + 16) * PD + d] = __logf(sgb + EPSF);
    }
    __syncthreads();

    if (t < ONC) {
      float L = 0.f;
#pragma unroll
      for (int dd = 0; dd < PD; ++dd) L += red_s[t * PD + dd];
      p0_s[t] = -L;                                  // p0 = -sum log(sigma+eps)
      float lh = rsum_s[t] * ((float)PD * bv[t] + L);// sum_d (bv+log sig)*rsum
      float z  = lam_arr[it] * (ba[t] - lh);
      actp_s[t] = 1.0f / (1.0f + __expf(-z));
    }
    __syncthreads();

    if (it < 2) {
      // ---- e_step: p1 + softmax over o (o = lane, full-wave shfl) ----
      const int o = lane;
      const float lact = __logf(actp_s[o] + EPSF) + p0_s[o];
      for (int j = 0; j < 36; ++j) {
        const int i = wave + 8 * j;
        // stage w[i,:,:] (512 f32) coalesced into this wave's padded LDS slot
#pragma unroll
        for (int k = 0; k < 16; ++k) {
          int m = lane + 32 * k;
          w_s[wave * 544 + m + (m >> 4)] = w[i * 512 + m];
        }
        __builtin_amdgcn_wave_barrier();
        float acc = 0.f;
#pragma unroll
        for (int dd = 0; dd < PD; ++dd) {
          int g  = i * PD + dd;
          int c  = (int)((unsigned)g / 9u);
          int ij = g - c * 9;
          float x  = x_s[ij * 512 + c];
          float v  = w_s[wave * 544 + o * 17 + dd] * x;
          float dm = v - mu_s[o * PD + dd];
          float sg = sig_s[o * PD + dd];
          acc += dm * dm / (2.f * sg * sg + EPSF);
        }
        float logit = lact - acc;
        float mx = logit;
#pragma unroll
        for (int off = 16; off > 0; off >>= 1) mx = fmaxf(mx, __shfl_xor(mx, off, 32));
        float e  = __expf(logit - mx);
        float se = e;
#pragma unroll
        for (int off = 16; off > 0; off >>= 1) se += __shfl_xor(se, off, 32);
        float r = e / se;
        rp_s[i * ONC + o] = r * a_s[(i % 9) * NC + (i / 9)];
        __builtin_amdgcn_wave_barrier();
      }
      __syncthreads();
    }
  }

  // outputs: pose_out = mu (288 x 512, channel = o*16+d), act_out = act_p
  float* pose_out = out;
  float* act_out  = out + (NB * OHW * OHW * 512);
  pose_out[pos * 512 + t]       = mu_s[t];
  pose_out[pos * 512 + 256 + t] = mu_s[256 + t];
  if (t < ONC) act_out[pos * ONC + t] = actp_s[t];
}

extern "C" void kernel_launch(void* const* d_in, const int* in_sizes, int n_in,
                              void* d_out, int out_size, void* d_ws, size_t ws_size,
                              hipStream_t stream) {
  const float* pose   = (const float*)d_in[0];
  const float* active = (const float*)d_in[1];
  const float* w      = (const float*)d_in[2];
  const float* bv     = (const float*)d_in[3];
  const float* ba     = (const float*)d_in[4];
  float* out = (float*)d_out;
  convcaps_em_kernel<<<dim3(NB * OHW * OHW), dim3(256), 0, stream>>>(
      pose, active, w, bv, ba, out);
}